// MiniDeepSeekV3Gate_67370857005272
// MI455X (gfx1250) — compile-verified
//
#include <hip/hip_runtime.h>

typedef float v2f __attribute__((ext_vector_type(2)));
typedef float v8f __attribute__((ext_vector_type(8)));

typedef int v4i_gcc __attribute__((vector_size(16)));
typedef __attribute__((address_space(1))) v4i_gcc* gvp_t;   // global int4*
typedef __attribute__((address_space(3))) v4i_gcc* lvp_t;   // LDS int4*

#define T_TOKENS 16384
#define HID      2048
#define NEXP     256
#define NGRP     8
#define EPG      32      // experts per group
#define TOPKG    4
#define TOPK     8
#define RSCALE   2.5f

#define TOK      128     // tokens per workgroup (amortizes weight re-reads from L2)
#define KC       16      // K chunk per LDS stage
#define NCHUNK   (HID / KC)   // 128
#define APITCH   20      // LDS pitch (floats): 16B-aligned float4 slots, bank-conflict-free
#define BPITCH   20
#define SPITCH   257     // score-tile pitch

// --- LDS float offsets: double-buffered stages; topk scratch aliases them ---
#define OFF_A0   0
#define OFF_A1   (TOK * APITCH)                    // 2560
#define OFF_B0   (2 * TOK * APITCH)                // 5120
#define OFF_B1   (OFF_B0 + NEXP * BPITCH)          // 10240
#define SMEM_FLOATS (OFF_B0 + 2 * NEXP * BPITCH)   // 15360 floats = 60 KB
// topk scratch (aliased, per 32-token batch):
#define OFF_SC   0                                  // 32*257      = 8224 floats
#define OFF_GV   (32 * SPITCH)                      // 32*8*8      = 2048 floats
#define OFF_GI   (OFF_GV + 32 * NGRP * TOPK)        // 2048 ints
#define OFF_GSUM (OFF_GI + 32 * NGRP * TOPK)        // 32*8 floats; end = 12576 < 15360

__global__ __launch_bounds__(256) void gate_fused_kernel(
    const float* __restrict__ x,     // (T, H)
    const float* __restrict__ w,     // (E, H)
    const float* __restrict__ bias,  // (E)
    float* __restrict__ wout,        // (T, 8)
    int*   __restrict__ iout)        // (T, 8)
{
    __shared__ float smem[SMEM_FLOATS];
    __shared__ float bias_s[NEXP];

    const int tid  = threadIdx.x;
    const int wave = tid >> 5;
    const int lane = tid & 31;
    const int tokBase = blockIdx.x * TOK;

    bias_s[tid] = bias[tid];   // blockDim == NEXP == 256

    // wave tile: 2 M-tiles x 8 N-tiles (32 tokens x 128 experts)
    const int mBase = (wave & 3) * 32;     // token sub-block within WG
    const int nBase = (wave >> 2) * 128;   // expert half

    v8f acc[2][8] = {};

    // Async global->LDS stage copy: no data VGPRs, tracked by ASYNCcnt.
    auto issue_stage = [&](int chunk, int buf) {
        const int k0 = chunk * KC;
        // A stage: 128 rows x 16 K = 512 float4; 2 per thread
        {
            const int row = tid >> 2;            // 0..63
            const int k   = (tid & 3) * 4;       // 0,4,8,12
            const int ab  = buf ? OFF_A1 : OFF_A0;
            __builtin_amdgcn_global_load_async_to_lds_b128(
                (gvp_t)(uintptr_t)&x[(size_t)(tokBase + row) * HID + k0 + k],
                (lvp_t)&smem[ab + row * APITCH + k], 0, 0);
            __builtin_amdgcn_global_load_async_to_lds_b128(
                (gvp_t)(uintptr_t)&x[(size_t)(tokBase + row + 64) * HID + k0 + k],
                (lvp_t)&smem[ab + (row + 64) * APITCH + k], 0, 0);
        }
        // B stage: 256 experts x 16 K = 1024 float4; 4 per thread
        {
            const int bo = buf ? OFF_B1 : OFF_B0;
            #pragma unroll
            for (int i = 0; i < 4; ++i) {
                const int f = i * 1024 + tid * 4;
                const int e = f >> 4;
                const int k = f & 15;
                __builtin_amdgcn_global_load_async_to_lds_b128(
                    (gvp_t)(uintptr_t)&w[(size_t)e * HID + k0 + k],
                    (lvp_t)&smem[bo + e * BPITCH + k], 0, 0);
            }
        }
    };

    issue_stage(0, 0);
    asm volatile("s_wait_asynccnt 0x0" ::: "memory");
    __syncthreads();

    const int off  = (lane >> 4) * 2;   // K-pair offset per ISA f32 A/B layout
    const int lrow = lane & 15;

    for (int c = 0; c < NCHUNK; ++c) {
        const int cur = c & 1;
        if (c + 1 < NCHUNK) issue_stage(c + 1, cur ^ 1);  // fire-and-forget prefetch
        const int ao = cur ? OFF_A1 : OFF_A0;
        const int bo = cur ? OFF_B1 : OFF_B0;
        #pragma unroll
        for (int kk = 0; kk < KC; kk += 4) {
            v2f a0 = *reinterpret_cast<const v2f*>(&smem[ao + (mBase +      lrow) * APITCH + kk + off]);
            v2f a1 = *reinterpret_cast<const v2f*>(&smem[ao + (mBase + 16 + lrow) * APITCH + kk + off]);
            #pragma unroll
            for (int nt = 0; nt < 8; ++nt) {
                v2f bf = *reinterpret_cast<const v2f*>(
                    &smem[bo + (nBase + nt * 16 + lrow) * BPITCH + kk + off]);
                acc[0][nt] = __builtin_amdgcn_wmma_f32_16x16x4_f32(
                    false, a0, false, bf, (short)0, acc[0][nt], false, false);
                acc[1][nt] = __builtin_amdgcn_wmma_f32_16x16x4_f32(
                    false, a1, false, bf, (short)0, acc[1][nt], false, false);
            }
        }
        asm volatile("s_wait_asynccnt 0x0" ::: "memory");  // prefetch landed (hidden by compute)
        __syncthreads();
    }

    // ---- routing in 4 batches of 32 tokens (score tile aliases dead stages) ----
    int*   gi_s   = reinterpret_cast<int*>(&smem[OFF_GI]);
    const int t   = tid >> 3;   // local token 0..31
    const int g   = tid & 7;    // group 0..7

    for (int b = 0; b < 4; ++b) {
        // waves owning this token batch write sigmoid scores to LDS
        if ((wave & 3) == b) {
            #pragma unroll
            for (int mt = 0; mt < 2; ++mt) {
                #pragma unroll
                for (int nt = 0; nt < 8; ++nt) {
                    #pragma unroll
                    for (int j = 0; j < 8; ++j) {
                        const int m = mt * 16 + j + 8 * (lane >> 4);   // C layout
                        const int e = nBase + nt * 16 + lrow;
                        const float s = acc[mt][nt][j];
                        smem[OFF_SC + m * SPITCH + e] = 1.0f / (1.0f + __expf(-s));
                    }
                }
            }
        }
        __syncthreads();

        // stage 1: 8 threads/token -> per-group sorted top-8 of (sigmoid + bias)
        {
            const float* sc = &smem[OFF_SC + t * SPITCH + g * EPG];
            const float* bs = &bias_s[g * EPG];
            unsigned taken = 0;
            float v0 = 0.f, v1 = 0.f;
            #pragma unroll
            for (int k = 0; k < TOPK; ++k) {
                float best = -INFINITY; int bi = 0; bool found = false;
                for (int i = 0; i < EPG; ++i) {
                    if ((taken >> i) & 1) continue;
                    const float v = sc[i] + bs[i];
                    if (!found || v > best) { best = v; bi = i; found = true; }
                }
                taken |= 1u << bi;
                smem[OFF_GV + (t * NGRP + g) * TOPK + k] = best;
                gi_s[(t * NGRP + g) * TOPK + k] = g * EPG + bi;
                if (k == 0) v0 = best;
                if (k == 1) v1 = best;
            }
            smem[OFF_GSUM + t * NGRP + g] = v0 + v1;   // group score = top-2 sum
        }
        __syncthreads();

        // stage 2: one thread/token -> top-4 groups, merge sorted lists, normalize
        if (g == 0) {
            const float* gsum = &smem[OFF_GSUM + t * NGRP];
            unsigned selm = 0;
            #pragma unroll
            for (int k = 0; k < TOPKG; ++k) {
                float best = -INFINITY; int bg = 0; bool found = false;
                for (int gg = 0; gg < NGRP; ++gg) {
                    if ((selm >> gg) & 1) continue;
                    if (!found || gsum[gg] > best) { best = gsum[gg]; bg = gg; found = true; }
                }
                selm |= 1u << bg;
            }
            int head[NGRP] = {0,0,0,0,0,0,0,0};
            int   idx[TOPK];
            float wv[TOPK];
            float wsum = 0.0f;
            const float* gv = &smem[OFF_GV + t * NGRP * TOPK];
            const int*   gi = &gi_s[t * NGRP * TOPK];
            const float* sc = &smem[OFF_SC + t * SPITCH];
            #pragma unroll
            for (int k = 0; k < TOPK; ++k) {
                float best = -INFINITY; int bg = 0; bool found = false;
                for (int gg = 0; gg < NGRP; ++gg) {       // ascending: ties -> lowest index
                    if (!((selm >> gg) & 1)) continue;
                    const float hv = gv[gg * TOPK + head[gg]];
                    if (!found || hv > best) { best = hv; bg = gg; found = true; }
                }
                const int e = gi[bg * TOPK + head[bg]];
                head[bg]++;
                idx[k] = e;
                wv[k]  = sc[e];     // ORIGINAL sigmoid score (no bias)
                wsum  += sc[e];
            }
            const float inv = RSCALE / (wsum + 1e-6f);
            const int token = tokBase + b * 32 + t;
            #pragma unroll
            for (int k = 0; k < TOPK; ++k) {
                wout[token * TOPK + k] = wv[k] * inv;
                iout[token * TOPK + k] = idx[k];
            }
        }
        __syncthreads();
    }
}

extern "C" void kernel_launch(void* const* d_in, const int* in_sizes, int n_in,
                              void* d_out, int out_size, void* d_ws, size_t ws_size,
                              hipStream_t stream) {
    (void)in_sizes; (void)n_in; (void)out_size; (void)d_ws; (void)ws_size;
    const float* x = (const float*)d_in[0];   // hidden_states (16384, 2048) f32
    const float* w = (const float*)d_in[1];   // weight        (256, 2048)  f32
    const float* b = (const float*)d_in[2];   // bias          (256)        f32
    float* wout = (float*)d_out;                                  // (16384, 8) f32
    int*   iout = (int*)(wout + (size_t)T_TOKENS * TOPK);         // (16384, 8) i32

    dim3 grid(T_TOKENS / TOK);   // 128 workgroups ~= one per WGP
    dim3 block(256);             // 8 wave32s
    gate_fused_kernel<<<grid, block, 0, stream>>>(x, w, b, wout, iout);
}